// Patcher_14525579395107
// MI455X (gfx1250) — compile-verified
//
#include <hip/hip_runtime.h>
#include <cstdint>

#define BB  16
#define HH  768
#define WW  768
#define CC  3
#define PHH 128
#define PWW 128
#define NBB 8
#define EPSF 1e-6f

// ---------------------------------------------------------------------------
// Kernel 1: bulk copy images -> out (float4 vectorized; this is ~226 MB of the
// ~241 MB total traffic, pure HBM bandwidth).
// ---------------------------------------------------------------------------
__global__ __launch_bounds__(256) void copy_images_kernel(
    const float4* __restrict__ src, float4* __restrict__ dst, int n4) {
  int i = blockIdx.x * blockDim.x + threadIdx.x;
  if (i < n4) dst[i] = src[i];
}

// ---------------------------------------------------------------------------
// Kernel 2: one workgroup per image; sequentially paste NB patches.
//  - stage 128x128x3 f32 patch (196.6 KB) into LDS via async global->LDS DMA
//  - per patch: wave32-shuffle + ds_add_f32 reduction of region stats from the
//    *current* image, then bilinear resample from LDS + affine hist-match.
// ---------------------------------------------------------------------------
__global__ __launch_bounds__(256, 1) void paste_kernel(
    const float* __restrict__ patch, const float* __restrict__ boxes,
    float* __restrict__ out) {
  __shared__ float spatch[PHH * PWW * CC];   // 196608 B (<=320KB/WGP on CDNA5)
  __shared__ float sacc[8];                  // [0..2]=sum, [4..6]=sumsq

  const int b   = blockIdx.x;
  const int tid = threadIdx.x;
  float* img = out + (size_t)b * (HH * WW * CC);

  // ---- Stage source patch into LDS with gfx1250 async loads (ASYNCcnt). ----
  for (int i = tid; i < (PHH * PWW * CC) / 4; i += 256) {
    unsigned lds_addr = (unsigned)(unsigned long long)(const void*)&spatch[i * 4];
    const float* g = patch + i * 4;
    asm volatile("global_load_async_to_lds_b128 %0, %1, off"
                 :: "v"(lds_addr), "v"(g) : "memory");
  }
#if __has_builtin(__builtin_amdgcn_s_wait_asynccnt)
  __builtin_amdgcn_s_wait_asynccnt(0);
#else
  asm volatile("s_wait_asynccnt 0" ::: "memory");
#endif
  __syncthreads();

  // ---- Patch statistics (constant across all pastes): mean, std+eps. ----
  if (tid < 8) sacc[tid] = 0.0f;
  __syncthreads();
  {
    float s[3] = {0.f, 0.f, 0.f}, q[3] = {0.f, 0.f, 0.f};
    for (int p = tid; p < PHH * PWW; p += 256) {
      const float* px = &spatch[p * 3];
#pragma unroll
      for (int c = 0; c < 3; ++c) { float v = px[c]; s[c] += v; q[c] += v * v; }
    }
#pragma unroll
    for (int off = 16; off > 0; off >>= 1) {
#pragma unroll
      for (int c = 0; c < 3; ++c) {
        s[c] += __shfl_xor(s[c], off, 32);
        q[c] += __shfl_xor(q[c], off, 32);
      }
    }
    if ((tid & 31) == 0) {
#pragma unroll
      for (int c = 0; c < 3; ++c) {
        atomicAdd(&sacc[c], s[c]);
        atomicAdd(&sacc[4 + c], q[c]);
      }
    }
  }
  __syncthreads();
  float pmean[3], pstd[3];
  {
    const float invN = 1.0f / (float)(PHH * PWW);
#pragma unroll
    for (int c = 0; c < 3; ++c) {
      pmean[c] = sacc[c] * invN;
      float var = sacc[4 + c] * invN - pmean[c] * pmean[c];
      pstd[c] = sqrtf(fmaxf(var, 0.0f)) + EPSF;   // jnp: std(...) + eps
    }
  }

  // ---- Sequential pastes (order matters: bg stats see previous pastes). ----
  for (int pi = 0; pi < NBB; ++pi) {
    const float* bx = boxes + ((size_t)b * NBB + pi) * 4;
    float ymin = bx[0], xmin = bx[1], ymax = bx[2], xmax = bx[3];
    float h = ymax - ymin, w = xmax - xmin;
    float pwf = h * 0.3f;
    float phf = pwf;                       // ASPECT == 1.0
    if (!(phf > 60.0f)) continue;          // uniform branch: valid filter

    float oy = ymin + h * 0.5f, ox = xmin + w * 0.5f;
    float y0 = fmaxf(oy - phf * 0.5f, 0.0f);
    float x0 = fmaxf(ox - pwf * 0.5f, 0.0f);
    if (y0 + phf > (float)HH) y0 = (float)HH - phf;
    if (x0 + pwf > (float)WW) x0 = (float)WW - pwf;
    int y0i = (int)y0, x0i = (int)x0;      // astype(int32): trunc (>=0)
    int phi = (int)phf, pwi = (int)pwf;
    int npix = phi * pwi;
    float cnt = (float)(npix > 0 ? npix : 1);

    __syncthreads();                       // prev iter's sacc reads complete
    if (tid < 8) sacc[tid] = 0.0f;
    __syncthreads();

    // pass 1: background stats over the current image region
    {
      float s[3] = {0.f, 0.f, 0.f}, q[3] = {0.f, 0.f, 0.f};
      for (int idx = tid; idx < npix; idx += 256) {
        int ry = idx / pwi, rx = idx - ry * pwi;
        size_t o = ((size_t)(y0i + ry) * WW + (x0i + rx)) * CC;
#pragma unroll
        for (int c = 0; c < 3; ++c) { float v = img[o + c]; s[c] += v; q[c] += v * v; }
      }
#pragma unroll
      for (int off = 16; off > 0; off >>= 1) {
#pragma unroll
        for (int c = 0; c < 3; ++c) {
          s[c] += __shfl_xor(s[c], off, 32);
          q[c] += __shfl_xor(q[c], off, 32);
        }
      }
      if ((tid & 31) == 0) {
#pragma unroll
        for (int c = 0; c < 3; ++c) {
          atomicAdd(&sacc[c], s[c]);
          atomicAdd(&sacc[4 + c], q[c]);
        }
      }
    }
    __syncthreads();                       // stats ready; region reads done

    float scale[3], offs[3];
#pragma unroll
    for (int c = 0; c < 3; ++c) {
      float m = sacc[c] / cnt;
      float bstd = sqrtf(sacc[4 + c] / cnt - m * m + EPSF);
      scale[c] = bstd / pstd[c];
      offs[c]  = m - pmean[c] * scale[c];  // matched = patch*scale + offs
    }

    // pass 2: bilinear resample (half-pixel centers) from LDS patch, write
    float syk = (float)PHH / fmaxf((float)phi, 1.0f);
    float sxk = (float)PWW / fmaxf((float)pwi, 1.0f);
    for (int idx = tid; idx < npix; idx += 256) {
      int ry = idx / pwi, rx = idx - ry * pwi;
      float sy = fminf(fmaxf(((float)ry + 0.5f) * syk - 0.5f, 0.0f), (float)(PHH - 1));
      float sx = fminf(fmaxf(((float)rx + 0.5f) * sxk - 0.5f, 0.0f), (float)(PWW - 1));
      int yi0 = (int)sy; float fy = sy - (float)yi0; int yi1 = min(yi0 + 1, PHH - 1);
      int xi0 = (int)sx; float fx = sx - (float)xi0; int xi1 = min(xi0 + 1, PWW - 1);
      const float* p00 = &spatch[(yi0 * PWW + xi0) * CC];
      const float* p01 = &spatch[(yi0 * PWW + xi1) * CC];
      const float* p10 = &spatch[(yi1 * PWW + xi0) * CC];
      const float* p11 = &spatch[(yi1 * PWW + xi1) * CC];
      size_t o = ((size_t)(y0i + ry) * WW + (x0i + rx)) * CC;
#pragma unroll
      for (int c = 0; c < 3; ++c) {
        float top = p00[c] + (p01[c] - p00[c]) * fx;
        float bot = p10[c] + (p11[c] - p10[c]) * fx;
        float v   = top + (bot - top) * fy;
        img[o + c] = v * scale[c] + offs[c];
      }
    }
    __threadfence();                       // writes visible to next iter reads
  }
}

// ---------------------------------------------------------------------------
extern "C" void kernel_launch(void* const* d_in, const int* in_sizes, int n_in,
                              void* d_out, int out_size, void* d_ws, size_t ws_size,
                              hipStream_t stream) {
  const float* images = (const float*)d_in[0];   // [B,H,W,C]
  const float* patch  = (const float*)d_in[1];   // [PH,PW,C]
  const float* boxes  = (const float*)d_in[2];   // [B,NB,4]
  float* out = (float*)d_out;

  const int n4 = (BB * HH * WW * CC) / 4;        // exact multiple of 4
  copy_images_kernel<<<(n4 + 255) / 256, 256, 0, stream>>>(
      (const float4*)images, (float4*)out, n4);
  paste_kernel<<<BB, 256, 0, stream>>>(patch, boxes, out);
}